// MiniModelBank_75952201662947
// MI455X (gfx1250) — compile-verified
//
#include <hip/hip_runtime.h>
#include <hip/hip_bf16.h>

typedef float v2f __attribute__((ext_vector_type(2)));
typedef float v8f __attribute__((ext_vector_type(8)));

__device__ __forceinline__ float wave_max(float v) {
#pragma unroll
  for (int m = 16; m >= 1; m >>= 1) v = fmaxf(v, __shfl_xor(v, m, 32));
  return v;
}
__device__ __forceinline__ float wave_sum(float v) {
#pragma unroll
  for (int m = 16; m >= 1; m >>= 1) v += __shfl_xor(v, m, 32);
  return v;
}

// One wave32 per model. 8 waves per 256-thread block.
__global__ __launch_bounds__(256)
void minibank_kernel(const float* __restrict__ C, const float* __restrict__ W1,
                     const float* __restrict__ b1, const float* __restrict__ Wp,
                     const float* __restrict__ bp, float* __restrict__ out,
                     int nModels, float scale) {
  __shared__ float sy[8][64];               // per-wave y scatter buffer (2 KB)
  const int lane  = threadIdx.x & 31;
  const int wave  = threadIdx.x >> 5;
  const int model = blockIdx.x * 8 + wave;
  if (model >= nModels) return;             // wave-uniform: EXEC stays all-1s

  const int h   = lane >> 4;                // half-wave: 0 or 1
  const int m16 = lane & 15;

  // ---- Step A: softmax(scale * C[model]) over 64 elems (2 per lane) ----
  const float* Cn = C + (size_t)model * 64;
  float c0 = Cn[lane], c1 = Cn[lane + 32];
  float mx = wave_max(fmaxf(c0, c1));
  float e0 = __expf(scale * (c0 - mx));
  float e1 = __expf(scale * (c1 - mx));
  float inv = 1.0f / wave_sum(e0 + e1);
  float x0 = e0 * inv, x1 = e1 * inv;       // lane holds x[lane], x[lane+32]

  // ---- Build B fragments for all 16 k-chunks (4x16, columns replicated) ----
  // B VGPR0: lanes 0-15 = row K=0, lanes 16-31 = row K=2; VGPR1: K=1 / K=3.
  v2f bmat[16];
  const int h2 = h << 1;
#pragma unroll
  for (int kc = 0; kc < 8; ++kc) {
    bmat[kc].x     = __shfl(x0, kc * 4 + h2,     32);
    bmat[kc].y     = __shfl(x0, kc * 4 + h2 + 1, 32);
    bmat[kc + 8].x = __shfl(x1, kc * 4 + h2,     32);
    bmat[kc + 8].y = __shfl(x1, kc * 4 + h2 + 1, 32);
  }

  // ---- Step B: y = W1[model] @ x via V_WMMA_F32_16X16X4_F32 ----
  // A 16x4 tile: lane (m16,h) holds W[16r+m16][4kc+2h], W[16r+m16][4kc+2h+1]
  // -> one b64 load per lane per tile; each lane streams 128B of its row.
  const float* Wn = W1 + (size_t)model * 4096;
#pragma unroll
  for (int r = 0; r < 4; ++r) {
    const float* ap = Wn + (size_t)(r * 16 + m16) * 64 + h2;
    v8f acc = {0.f, 0.f, 0.f, 0.f, 0.f, 0.f, 0.f, 0.f};
#pragma unroll
    for (int kc = 0; kc < 16; ++kc) {
      v2f a = *(const v2f*)(ap + kc * 4);
      acc = __builtin_amdgcn_wmma_f32_16x16x4_f32(
          /*neg_a=*/false, a, /*neg_b=*/false, bmat[kc],
          /*c_mod=*/(short)0, acc, /*reuse_a=*/false, /*reuse_b=*/false);
    }
    // D: VGPR j holds row 16r + 8h + j (all 16 columns identical).
    // Scatter one column per half-wave into LDS.
    if (m16 == 0) {
#pragma unroll
      for (int j = 0; j < 8; ++j) sy[wave][r * 16 + h * 8 + j] = acc[j];
    }
  }
  __builtin_amdgcn_wave_barrier();
  asm volatile("s_wait_dscnt 0" ::: "memory");

  // ---- bias + relu, back to 2-elems-per-lane layout ----
  const float* b1n = b1 + (size_t)model * 64;
  float y0 = fmaxf(sy[wave][lane]      + b1n[lane],      0.0f);
  float y1 = fmaxf(sy[wave][lane + 32] + b1n[lane + 32], 0.0f);

  // ---- head: 2x64 linear + softmax over 2 ----
  const float* Wpn = Wp + (size_t)model * 128;
  float l0 = wave_sum(Wpn[lane] * y0 + Wpn[lane + 32] * y1) + bp[model * 2 + 0];
  float l1 = wave_sum(Wpn[64 + lane] * y0 + Wpn[96 + lane] * y1) + bp[model * 2 + 1];
  float lm = fmaxf(l0, l1);
  float p0 = __expf(l0 - lm), p1 = __expf(l1 - lm);
  float pn = 1.0f / (p0 + p1);
  p0 *= pn; p1 *= pn;

  // ---- Step C: tanh combine, coalesced store ----
  float* On = out + (size_t)model * 64;
  On[lane]      = tanhf(p0 * y0) + tanhf(p1 * y0);
  On[lane + 32] = tanhf(p0 * y1) + tanhf(p1 * y1);
}

extern "C" void kernel_launch(void* const* d_in, const int* in_sizes, int n_in,
                              void* d_out, int out_size, void* d_ws, size_t ws_size,
                              hipStream_t stream) {
  const float* C  = (const float*)d_in[0];
  const float* W1 = (const float*)d_in[1];
  const float* b1 = (const float*)d_in[2];
  const float* Wp = (const float*)d_in[3];
  const float* bp = (const float*)d_in[4];
  float* out = (float*)d_out;
  const int n = in_sizes[0] / 64;           // N models
  const float scale = (float)n;             // n_pers
  const int blocks = (n + 7) / 8;           // 8 waves (models) per block
  minibank_kernel<<<blocks, 256, 0, stream>>>(C, W1, b1, Wp, bp, out, n, scale);
}